// GeneralContactLoss_47519518163567
// MI455X (gfx1250) — compile-verified
//
#include <hip/hip_runtime.h>

// Problem constants from the reference
#define NV_I   10475
#define NVPAD  10480          // ceil(NV/16)*16
#define NTILE  655            // NVPAD/16
#define B_I    2
#define K_I    90
#define FL_I   1000
#define CW     1024           // column chunk staged in LDS (multiple of 16)
#define WN_TH  0.99f
#define INV2PI 0.15915494309189535f

typedef float v2f __attribute__((ext_vector_type(2)));
typedef float v8f __attribute__((ext_vector_type(8)));

#if defined(__AMDGCN__) && __has_builtin(__builtin_amdgcn_wmma_f32_16x16x4_f32)
#define USE_WMMA 1
#else
#define USE_WMMA 0
#endif

// ---------------- pack A-side (-2x,-2y,-2z,1), B-side (x,y,z,|y|^2), norms ---
__global__ __launch_bounds__(256)
void k_prep(const float* __restrict__ v1, const float* __restrict__ v2,
            float* __restrict__ XpA, float* __restrict__ XpB,
            float* __restrict__ N2) {
  int idx = blockIdx.x * blockDim.x + threadIdx.x;
  if (idx >= B_I * 2 * NVPAD) return;
  int i  = idx % NVPAD;
  int bs = idx / NVPAD;            // b*2 + s
  int s = bs & 1, b = bs >> 1;
  const float* v = s ? v2 : v1;
  float x, y, z;
  if (i < NV_I) {
    const float* p = v + ((size_t)b * NV_I + i) * 3;
    x = p[0]; y = p[1]; z = p[2];
  } else {                         // sentinel row: enormous distance, never the min
    x = 1.0e9f; y = 0.f; z = 0.f;
  }
  float n2 = x * x + y * y + z * z;
  float4 a; a.x = -2.f * x; a.y = -2.f * y; a.z = -2.f * z; a.w = 1.f;
  float4 bq; bq.x = x; bq.y = y; bq.z = z; bq.w = n2;
  ((float4*)XpA)[idx] = a;
  ((float4*)XpB)[idx] = bq;
  N2[idx] = n2;
}

// ---------------- close_mouth mean (1 wave per (b, which)) ----------------
__global__ void k_mouth(const float* __restrict__ v1, const float* __restrict__ v2,
                        const int* __restrict__ ids, float* __restrict__ mv,
                        unsigned* __restrict__ anyflag) {
  int bs = blockIdx.x;             // b*2 + w
  int w = bs & 1, b = bs >> 1;
  const float* v = (w ? v2 : v1) + (size_t)b * NV_I * 3;
  int lane = threadIdx.x;
  float sx = 0.f, sy = 0.f, sz = 0.f;
  for (int k = lane; k < K_I; k += 32) {
    int id = ids[k];
    sx += v[id * 3 + 0]; sy += v[id * 3 + 1]; sz += v[id * 3 + 2];
  }
  for (int m = 16; m >= 1; m >>= 1) {
    sx += __shfl_xor(sx, m, 32);
    sy += __shfl_xor(sy, m, 32);
    sz += __shfl_xor(sz, m, 32);
  }
  if (lane == 0) {
    const float inv = 1.0f / (float)K_I;
    mv[bs * 4 + 0] = sx * inv;
    mv[bs * 4 + 1] = sy * inv;
    mv[bs * 4 + 2] = sz * inv;
    mv[bs * 4 + 3] = 0.f;
    anyflag[bs] = 0u;              // zero-init the any(interior) flags each call
  }
}

// ---------------- gather low-res triangles ----------------
__global__ __launch_bounds__(256)
void k_tris(const float* __restrict__ v1, const float* __restrict__ v2,
            const float* __restrict__ mv, const int* __restrict__ vid,
            const int* __restrict__ faces, float* __restrict__ tris) {
  int t = blockIdx.x * blockDim.x + threadIdx.x;
  if (t >= B_I * 2 * FL_I) return;
  int f  = t % FL_I;
  int bs = t / FL_I;               // b*2 + w
  int w = bs & 1, b = bs >> 1;
  const float* v = (w ? v2 : v1) + (size_t)b * NV_I * 3;
  float* o = tris + (size_t)t * 9;
  for (int c = 0; c < 3; ++c) {
    int li = faces[f * 3 + c];
    int id = vid[li];
    if (id == NV_I) {              // mean ("mouth") vertex
      o[c*3+0] = mv[bs*4+0]; o[c*3+1] = mv[bs*4+1]; o[c*3+2] = mv[bs*4+2];
    } else {
      o[c*3+0] = v[id*3+0]; o[c*3+1] = v[id*3+1]; o[c*3+2] = v[id*3+2];
    }
  }
}

// ---------------- winding numbers (tris staged in LDS) ----------------
__global__ __launch_bounds__(256)
void k_wind(const float* __restrict__ XpB, const float* __restrict__ tris,
            float* __restrict__ IntF, unsigned* __restrict__ anyflag) {
  __shared__ float sT[FL_I * 9];   // 36 KB: all triangles of the opposite set
  int b = blockIdx.y, s = blockIdx.z;
  int bs = b * 2 + s;
  const float* tb = tris + (size_t)(b * 2 + (1 - s)) * FL_I * 9;
  for (int t = threadIdx.x; t < FL_I * 9; t += blockDim.x) sT[t] = tb[t];
  __syncthreads();
  int i = blockIdx.x * blockDim.x + threadIdx.x;
  if (i >= NVPAD) return;
  float flag = 0.f;
  if (i < NV_I) {
    const float* P = XpB + ((size_t)bs * NVPAD + i) * 4;
    float px = P[0], py = P[1], pz = P[2];
    float sum = 0.f;
    for (int f = 0; f < FL_I; ++f) {
      const float* tr = &sT[f * 9];
      float ax = tr[0]-px, ay = tr[1]-py, az = tr[2]-pz;
      float bx = tr[3]-px, by = tr[4]-py, bz = tr[5]-pz;
      float cx = tr[6]-px, cy = tr[7]-py, cz = tr[8]-pz;
      float la = sqrtf(ax*ax + ay*ay + az*az);
      float lb = sqrtf(bx*bx + by*by + bz*bz);
      float lc = sqrtf(cx*cx + cy*cy + cz*cz);
      float ux = by*cz - bz*cy;
      float uy = bz*cx - bx*cz;
      float uz = bx*cy - by*cx;
      float det = ax*ux + ay*uy + az*uz;
      float ab = ax*bx + ay*by + az*bz;
      float bc = bx*cx + by*cy + bz*cz;
      float ca = cx*ax + cy*ay + cz*az;
      float den = la*lb*lc + ab*lc + bc*la + ca*lb;
      sum += atan2f(det, den);
    }
    float wn = sum * INV2PI;       // sum(2*atan2)/(4*pi)
    if (wn >= WN_TH) { flag = 1.f; atomicOr(&anyflag[bs], 1u); }
  }
  IntF[(size_t)bs * NVPAD + i] = flag;   // 0 for padded rows
}

// ---------------- WMMA pairwise-min kernel ----------------
// A(16x4) = (-2x,-2y,-2z,1) rows, B(4x16) = (x,y,z,|y|^2) cols, C = 0
//   =>  D[i][j] = |y_j|^2 - 2 x_i.y_j      (the |y|^2 rides in the K=4 slot)
// min over j; |x_i|^2 and the 1e-12 clamp applied after the sweep.
__global__ __launch_bounds__(256)
void k_minpass(const float* __restrict__ XpA, const float* __restrict__ XpB,
               const float* __restrict__ N2, float* __restrict__ MinD2) {
  __shared__ __align__(16) float sY[CW * 4];   // 16 KB B-side coords+norm
  const int b = blockIdx.y, s = blockIdx.z;
  const int tid  = threadIdx.x;
  const int wave = tid >> 5, lane = tid & 31;
  const int n = lane & 15, hi = lane >> 4;
  const int rowTile = blockIdx.x * 8 + wave;
  const int i0 = rowTile * 16;
  const float* XA  = XpA + (size_t)(b*2 + s)       * NVPAD * 4;
  const float* YB  = XpB + (size_t)(b*2 + (1 - s)) * NVPAD * 4;
  const float* xn2 = N2  + (size_t)(b*2 + s)       * NVPAD;
  const bool act = rowTile < NTILE;

  // A fragment (16x4 f32, 2 VGPRs): lane<16 -> K=0,1 ; lane>=16 -> K=2,3
  v2f a2; a2.x = 0.f; a2.y = 0.f;
  float xs[8];
#if !USE_WMMA
  float xr[8][3];
#endif
  if (act) {
    a2.x = XA[(i0 + n) * 4 + 2*hi + 0];
    a2.y = XA[(i0 + n) * 4 + 2*hi + 1];
#pragma unroll
    for (int r = 0; r < 8; ++r) xs[r] = xn2[i0 + r + 8*hi];
#if !USE_WMMA
#pragma unroll
    for (int r = 0; r < 8; ++r) {
      xr[r][0] = YB[0];  // placeholder, overwritten below from own set
    }
#pragma unroll
    for (int r = 0; r < 8; ++r) {
      const float* xp = XpB + ((size_t)(b*2 + s) * NVPAD + i0 + r + 8*hi) * 4;
      xr[r][0] = xp[0]; xr[r][1] = xp[1]; xr[r][2] = xp[2];
    }
#endif
  } else {
#pragma unroll
    for (int r = 0; r < 8; ++r) xs[r] = 0.f;
  }

  float m[8];
#pragma unroll
  for (int r = 0; r < 8; ++r) m[r] = 3.4e38f;

#if USE_WMMA
  const v8f zeroC = {0.f, 0.f, 0.f, 0.f, 0.f, 0.f, 0.f, 0.f};
#endif

  for (int jb = 0; jb < NVPAD; jb += CW) {
    const int cw = (NVPAD - jb < CW) ? (NVPAD - jb) : CW;
    __syncthreads();
    for (int t = tid; t < cw; t += 256) {           // cooperative b128 stage
      ((float4*)sY)[t] = ((const float4*)YB)[jb + t];
      if (jb + CW < NVPAD)                          // prefetch next chunk
        __builtin_prefetch(&((const float4*)YB)[jb + CW + t], 0, 1);
    }
    __syncthreads();
    if (act) {                                      // wave-uniform: EXEC all-1s
      const int ntile = cw >> 4;
      for (int jt = 0; jt < ntile; ++jt) {
#if USE_WMMA
        // B fragment (4x16 f32): lane<16 -> K=0,1 ; lane>=16 -> K=2,3 (N=n)
        const float2 bv = *(const float2*)&sY[jt*64 + n*4 + 2*hi];
        v2f bf; bf.x = bv.x; bf.y = bv.y;
        v8f d = __builtin_amdgcn_wmma_f32_16x16x4_f32(
            false, a2, false, bf, (short)0, zeroC, false, false);
#pragma unroll
        for (int r = 0; r < 8; ++r) m[r] = fminf(m[r], d[r]);
#else
        const float bx = sY[jt*64 + n*4 + 0];
        const float by = sY[jt*64 + n*4 + 1];
        const float bz = sY[jt*64 + n*4 + 2];
        const float yn = sY[jt*64 + n*4 + 3];
#pragma unroll
        for (int r = 0; r < 8; ++r) {
          float d2p = yn - 2.f*(xr[r][0]*bx + xr[r][1]*by + xr[r][2]*bz);
          m[r] = fminf(m[r], d2p);
        }
#endif
      }
    }
  }
  if (act) {
#pragma unroll
    for (int r = 0; r < 8; ++r) {
      float v = m[r];
      for (int msk = 1; msk < 16; msk <<= 1)        // min across 16 lanes/half
        v = fminf(v, __shfl_xor(v, msk, 32));
      if (n == 0) {
        int row = i0 + r + 8*hi;
        if (row < NV_I)
          MinD2[(size_t)(b*2 + s) * NVPAD + row] = fmaxf(xs[r] + v, 1e-12f);
      }
    }
  }
}

// ---------------- gated reduction -> scalar ----------------
__global__ __launch_bounds__(256)
void k_final(const float* __restrict__ MinD2, const float* __restrict__ IntF,
             const unsigned* __restrict__ anyflag, float* __restrict__ out) {
  __shared__ float acc0[256], acc1[256];
  float a0 = 0.f, a1 = 0.f;
  for (int idx = threadIdx.x; idx < B_I * 2 * NVPAD; idx += 256) {
    float v = MinD2[idx] * 1.0e6f * IntF[idx];      // (min d * 1000)^2 gated
    if (idx < 2 * NVPAD) a0 += v; else a1 += v;
  }
  acc0[threadIdx.x] = a0; acc1[threadIdx.x] = a1;
  __syncthreads();
  for (int off = 128; off > 0; off >>= 1) {
    if (threadIdx.x < off) {
      acc0[threadIdx.x] += acc0[threadIdx.x + off];
      acc1[threadIdx.x] += acc1[threadIdx.x + off];
    }
    __syncthreads();
  }
  if (threadIdx.x == 0) {
    float b0 = (anyflag[0] && anyflag[1]) ? acc0[0] : 0.f;
    float b1 = (anyflag[2] && anyflag[3]) ? acc1[0] : 0.f;
    out[0] = (b0 + b1) * 0.5f;
  }
}

extern "C" void kernel_launch(void* const* d_in, const int* in_sizes, int n_in,
                              void* d_out, int out_size, void* d_ws, size_t ws_size,
                              hipStream_t stream) {
  const float* v1  = (const float*)d_in[0];
  const float* v2  = (const float*)d_in[1];
  const int* ids   = (const int*)d_in[2];
  const int* vid   = (const int*)d_in[3];
  const int* faces = (const int*)d_in[4];

  float* ws   = (float*)d_ws;                       // ~2.0 MB total
  float* XpA  = ws;                                 // B*2*NVPAD*4 floats
  float* XpB  = XpA + (size_t)B_I * 2 * NVPAD * 4;  // B*2*NVPAD*4
  float* N2   = XpB + (size_t)B_I * 2 * NVPAD * 4;  // B*2*NVPAD
  float* Mn   = N2  + (size_t)B_I * 2 * NVPAD;      // B*2*NVPAD
  float* In   = Mn  + (size_t)B_I * 2 * NVPAD;      // B*2*NVPAD
  float* mv   = In  + (size_t)B_I * 2 * NVPAD;      // 16
  float* tris = mv + 16;                            // B*2*FL*9
  unsigned* anyflag = (unsigned*)(tris + (size_t)B_I * 2 * FL_I * 9);  // 4

  k_prep <<<dim3((B_I*2*NVPAD + 255)/256), 256, 0, stream>>>(v1, v2, XpA, XpB, N2);
  k_mouth<<<dim3(B_I*2), 32, 0, stream>>>(v1, v2, ids, mv, anyflag);
  k_tris <<<dim3((B_I*2*FL_I + 255)/256), 256, 0, stream>>>(v1, v2, mv, vid, faces, tris);
  k_wind <<<dim3((NVPAD + 255)/256, B_I, 2), 256, 0, stream>>>(XpB, tris, In, anyflag);
  k_minpass<<<dim3((NTILE + 7)/8, B_I, 2), 256, 0, stream>>>(XpA, XpB, N2, Mn);
  k_final<<<1, 256, 0, stream>>>(Mn, In, anyflag, (float*)d_out);

  (void)in_sizes; (void)n_in; (void)out_size; (void)ws_size;
}